// SlotConditionedHead_19232863551762
// MI455X (gfx1250) — compile-verified
//
#include <hip/hip_runtime.h>
#include <cstdint>

typedef float v2f __attribute__((ext_vector_type(2)));
typedef float v8f __attribute__((ext_vector_type(8)));

#define B_   2
#define C_   32
#define S_   16
#define E_   64
#define DHW_ (96 * 96 * 96)          // 884736
#define TILE_P 128
#define LDS_STRIDE 34                // floats per [p] row; conflict-free b64 reads
#define TILES_PER_BATCH (DHW_ / TILE_P)  // 6912

#if __has_builtin(__builtin_amdgcn_global_load_async_to_lds_b32)
#define USE_ASYNC_LDS 1
typedef __attribute__((address_space(1))) int gint_t;
typedef __attribute__((address_space(3))) int lint_t;
#endif

// ---------------------------------------------------------------------------
// Kernel 1: tiny FiLM MLP -> eff_w (16x32), eff_b (16). One wave, 16 live lanes.
// ---------------------------------------------------------------------------
__global__ void film_prep_kernel(const float* __restrict__ se, const float* __restrict__ w1,
                                 const float* __restrict__ b1, const float* __restrict__ w2,
                                 const float* __restrict__ b2, const float* __restrict__ cw,
                                 const float* __restrict__ cb, float* __restrict__ effw,
                                 float* __restrict__ effb) {
  const int s = threadIdx.x;
  if (s >= S_) return;
  float h[E_];
  for (int e = 0; e < E_; ++e) {
    float acc = b1[e];
    for (int j = 0; j < E_; ++j) acc = fmaf(se[s * E_ + j], w1[e * E_ + j], acc);
    h[e] = fmaxf(acc, 0.0f);
  }
  float eb = cb[s];
  for (int c = 0; c < C_; ++c) {
    float sc = b2[c];
    float sh = b2[c + C_];
    for (int e = 0; e < E_; ++e) {
      sc = fmaf(h[e], w2[c * E_ + e], sc);
      sh = fmaf(h[e], w2[(c + C_) * E_ + e], sh);
    }
    const float w = cw[s * C_ + c];
    effw[s * C_ + c] = w * (1.0f + sc);   // conv_w * (1 + scale)
    eb = fmaf(w, sh, eb);                 // sum(conv_w * shift)
  }
  effb[s] = eb;
}

// ---------------------------------------------------------------------------
// Kernel 2: out[b,s,p] = sum_c X[b,c,p] * eff_w[s,c] + eff_b[s]
// Memory-bound streaming GEMM: M=16 (slots), K=32 (channels), N=positions.
// Block = 256 threads (8 waves); block tile = 128 positions; wave tile = 16.
// V_WMMA_F32_16X16X4_F32, 8 K-chunks of 4 channels.
// ---------------------------------------------------------------------------
__global__ __launch_bounds__(256) void film_einsum_kernel(
    const float* __restrict__ x, const float* __restrict__ effw,
    const float* __restrict__ effb, float* __restrict__ out) {
  __shared__ float xs[TILE_P * LDS_STRIDE];  // [p][c], padded: 17408 B

  const int tid  = threadIdx.x;
  const int lane = tid & 31;
  const int b    = blockIdx.x / TILES_PER_BATCH;
  const int p0   = (blockIdx.x % TILES_PER_BATCH) * TILE_P;

  // ---- Stage X[b, 0:32, p0:p0+128] into LDS, transposed to [p][c] ----
  // Global reads: 128B contiguous per (c,row); LDS writes: stride-34 -> all
  // 32 lanes of a wave land in 32 distinct even-residue banks (no conflicts).
  const int pl    = tid & (TILE_P - 1);  // 0..127
  const int chome = tid >> 7;            // 0 or 1
  const float* __restrict__ xb =
      x + (size_t)b * C_ * DHW_ + (size_t)chome * DHW_ + p0 + pl;
  float* __restrict__ ls = xs + pl * LDS_STRIDE + chome;
#pragma unroll
  for (int cc = 0; cc < C_; cc += 2) {
#ifdef USE_ASYNC_LDS
    __builtin_amdgcn_global_load_async_to_lds_b32(
        (gint_t*)(uintptr_t)(xb + (size_t)cc * DHW_),
        (lint_t*)(uint32_t)(uintptr_t)(ls + cc), /*offset=*/0, /*cpol=*/0);
#else
    ls[cc] = xb[(size_t)cc * DHW_];
#endif
  }
#ifdef USE_ASYNC_LDS
#if __has_builtin(__builtin_amdgcn_s_wait_asynccnt)
  __builtin_amdgcn_s_wait_asynccnt(0);
#else
  asm volatile("s_wait_asynccnt 0" ::: "memory");
#endif
#endif
  __syncthreads();

  // ---- Per-wave WMMA tile: D(16x16) = eff_w(16x32) x Xtile(32x16) + bias ----
  const int m    = lane & 15;      // A row / B column / output N index
  const int half = lane >> 4;      // 0: K=0,1 slice   1: K=2,3 slice
  const int khi  = half * 2;
  const int sb   = half * 8;       // C/D: VGPR r holds row (r + 8*half)

  // A fragments: ISA 16x4 f32 layout -> lane<16: {K0,K1}, lane>=16: {K2,K3}
  v2f a[8];
#pragma unroll
  for (int k = 0; k < 8; ++k)
    a[k] = *(const v2f*)(effw + m * C_ + k * 4 + khi);

  // Accumulator initialized with eff_b broadcast along N
  v8f acc;
#pragma unroll
  for (int r = 0; r < 8; ++r) acc[r] = effb[sb + r];

  const int wv = tid >> 5;
  const int pw = wv * 16 + m;      // this lane's position column in the tile
#pragma unroll
  for (int k = 0; k < 8; ++k) {
    // B fragment: lanes 0-15 get rows {4k,4k+1}, lanes 16-31 get {4k+2,4k+3}
    const v2f bf = *(const v2f*)(xs + pw * LDS_STRIDE + k * 4 + khi);
    acc = __builtin_amdgcn_wmma_f32_16x16x4_f32(false, a[k], false, bf,
                                                (short)0, acc, false, false);
  }

  // ---- Store: VGPR r -> out[b, sb+r, p0+pw] (64B contiguous per half-wave) ----
  float* __restrict__ op = out + ((size_t)b * S_ + sb) * DHW_ + p0 + pw;
#pragma unroll
  for (int r = 0; r < 8; ++r)
    op[(size_t)r * DHW_] = acc[r];
}

// ---------------------------------------------------------------------------
extern "C" void kernel_launch(void* const* d_in, const int* in_sizes, int n_in,
                              void* d_out, int out_size, void* d_ws, size_t ws_size,
                              hipStream_t stream) {
  (void)in_sizes; (void)n_in; (void)out_size; (void)ws_size;
  const float* x  = (const float*)d_in[0];  // shared_features (2,32,96,96,96)
  const float* se = (const float*)d_in[1];  // slot_embed (16,64)
  const float* w1 = (const float*)d_in[2];  // (64,64)
  const float* b1 = (const float*)d_in[3];  // (64,)
  const float* w2 = (const float*)d_in[4];  // (64,64)
  const float* b2 = (const float*)d_in[5];  // (64,)
  const float* cw = (const float*)d_in[6];  // conv_w (16,32)
  const float* cb = (const float*)d_in[7];  // conv_b (16,)
  float* out  = (float*)d_out;
  float* effw = (float*)d_ws;               // 512 floats
  float* effb = effw + S_ * C_;             // 16 floats

  film_prep_kernel<<<1, 32, 0, stream>>>(se, w1, b1, w2, b2, cw, cb, effw, effb);
  film_einsum_kernel<<<B_ * TILES_PER_BATCH, 256, 0, stream>>>(x, effw, effb, out);
}